// GGNN_32779190403505
// MI455X (gfx1250) — compile-verified
//
#include <hip/hip_runtime.h>
#include <math.h>

#define CCH 30          // channels
#define NLAYERS 3
#define AST 34          // padded LDS row stride (floats): bank-conflict-free b64 reads
#define CH2 (CCH / 2)   // 15 float2 per row

typedef __attribute__((ext_vector_type(2))) float v2f;
typedef __attribute__((ext_vector_type(8))) float v8f;

__device__ __forceinline__ float sigmoid_f(float x) { return 1.0f / (1.0f + __expf(-x)); }

// ---------------------------------------------------------------------------
// m = h @ W  (W is [30,30] row-major, in x out) via V_WMMA_F32_16X16X4_F32.
// Block = 128 threads (4 waves). W staged transposed + zero-padded in LDS;
// each wave stages its 16x30 A-tile zero-padded in LDS (float2, coalesced,
// guard-free on full tiles). Fragments are unconditional ds_load_b64.
// D is staged back through the (dead) A tile in LDS — padding absorbs the
// acc1 overflow columns — then copied out as coalesced float2 stores.
// ---------------------------------------------------------------------------
__global__ void ggnn_transform_wmma(const float* __restrict__ h,
                                    const float* __restrict__ W,
                                    float* __restrict__ m, int N) {
    __shared__ float s_Wt[32 * AST];      // s_Wt[c*AST + k] = W[k][c], padded
    __shared__ float s_A[4][16 * AST];    // per-wave A tile [r][k], padded

    const int tid  = threadIdx.x;
    const int lane = tid & 31;
    const int warp = tid >> 5;
    const int row0 = (blockIdx.x * 4 + warp) * 16;
    const bool full = (row0 + 16 <= N);   // wave-uniform fast path

    // zero-fill (covers K/col padding)
    for (int i = tid; i < 32 * AST; i += 128) s_Wt[i] = 0.0f;
    for (int i = lane; i < 16 * AST; i += 32) s_A[warp][i] = 0.0f;
    __syncthreads();

    // stage W transposed: coalesced read of W, scatter into LDS
    for (int i = tid; i < CCH * CCH; i += 128) {
        const int k = i / CCH, c = i % CCH;
        s_Wt[c * AST + k] = W[i];
    }
    // stage A tile: 16 rows x 30 cols
    if (full) {
        const float2* h2 = reinterpret_cast<const float2*>(h + (size_t)row0 * CCH);
        #pragma unroll
        for (int i = lane; i < 16 * CH2; i += 32) {
            const int r = i / CH2, c2 = i % CH2;
            *reinterpret_cast<float2*>(&s_A[warp][r * AST + 2 * c2]) =
                h2[(size_t)r * CH2 + c2];
        }
    } else {
        for (int i = lane; i < 16 * CCH; i += 32) {
            const int r = i / CCH, c = i % CCH;
            const int row = row0 + r;
            if (row < N) s_A[warp][r * AST + c] = h[(size_t)row * CCH + c];
        }
    }
    __syncthreads();                      // last block-wide barrier

    if (row0 >= N) return;                // wave-uniform

    const int half = lane >> 4;           // 0: lanes 0-15, 1: lanes 16-31
    const int l    = lane & 15;
    v8f acc0 = {};                        // cols 0..15
    v8f acc1 = {};                        // cols 16..31 (30,31 padded)
    float* sa = &s_A[warp][0];

    #pragma unroll
    for (int k0 = 0; k0 < 32; k0 += 4) {
        const int ka = k0 + half * 2;     // this lane's K pair (8B aligned)
        const v2f a  = *(const v2f*)&sa  [l        * AST + ka];
        const v2f b0 = *(const v2f*)&s_Wt[l        * AST + ka];
        const v2f b1 = *(const v2f*)&s_Wt[(16 + l) * AST + ka];
        acc0 = __builtin_amdgcn_wmma_f32_16x16x4_f32(false, a, false, b0,
                                                     (short)0, acc0, false, false);
        acc1 = __builtin_amdgcn_wmma_f32_16x16x4_f32(false, a, false, b1,
                                                     (short)0, acc1, false, false);
    }

    // D layout: VGPR v -> row (v + half*8), col = l (+16 for acc1).
    // Stage through LDS (wave-local, in-order DS): cols 30,31 land in padding.
    #pragma unroll
    for (int v = 0; v < 8; ++v) {
        const int r = v + half * 8;
        sa[r * AST + l]      = acc0[v];
        sa[r * AST + 16 + l] = acc1[v];   // unconditional: stride 34 > 32
    }

    // coalesced copy-out
    if (full) {
        float2* m2 = reinterpret_cast<float2*>(m + (size_t)row0 * CCH);
        #pragma unroll
        for (int i = lane; i < 16 * CH2; i += 32) {
            const int r = i / CH2, c2 = i % CH2;
            m2[(size_t)r * CH2 + c2] =
                *reinterpret_cast<const float2*>(&sa[r * AST + 2 * c2]);
        }
    } else {
        for (int i = lane; i < 16 * CCH; i += 32) {
            const int r = i / CCH, c = i % CCH;
            const int row = row0 + r;
            if (row < N) m[(size_t)row * CCH + c] = sa[r * AST + c];
        }
    }
}

// ---------------------------------------------------------------------------
__global__ void ggnn_copy(const float* __restrict__ src, float* __restrict__ dst,
                          int n) {
    const int n4 = n >> 2;
    int i = blockIdx.x * blockDim.x + threadIdx.x;
    if (i < n4) {
        reinterpret_cast<float4*>(dst)[i] =
            reinterpret_cast<const float4*>(src)[i];
    }
    if (i < (n & 3)) {                     // scalar tail
        const int t = n4 * 4 + i;
        dst[t] = src[t];
    }
}

__global__ void ggnn_zero(float* __restrict__ p, int n) {
    const int n4 = n >> 2;
    int i = blockIdx.x * blockDim.x + threadIdx.x;
    if (i < n4) reinterpret_cast<float4*>(p)[i] = make_float4(0.f, 0.f, 0.f, 0.f);
    if (i < (n & 3)) p[n4 * 4 + i] = 0.0f;
}

// ---------------------------------------------------------------------------
// agg[dst[e]] += m[src[e]]  — thread per edge. Row gather as 15x float2
// (rows are 120 B, 8 B aligned), then 30 f32 atomics resolving in L2
// (agg is 24 MB -> resident in the 192 MB L2).
// ---------------------------------------------------------------------------
__global__ void ggnn_scatter(const int* __restrict__ src,
                             const int* __restrict__ dst,
                             const float* __restrict__ m,
                             float* __restrict__ agg, int E) {
    int e = blockIdx.x * blockDim.x + threadIdx.x;
    if (e >= E) return;
    const int s = src[e];
    const int d = dst[e];
    const float2* ms = reinterpret_cast<const float2*>(m + (size_t)s * CCH);
    float*        ad = agg + (size_t)d * CCH;

    float2 v[CH2];
    #pragma unroll
    for (int i = 0; i < CH2; ++i) v[i] = ms[i];
    #pragma unroll
    for (int i = 0; i < CH2; ++i) {
        atomicAdd(&ad[2 * i],     v[i].x);
        atomicAdd(&ad[2 * i + 1], v[i].y);
    }
}

// ---------------------------------------------------------------------------
// Fused GRU cell, thread per node, weights staged in LDS (broadcast reads).
// h updated in place (node-local; old row cached in registers).
// ---------------------------------------------------------------------------
__global__ void ggnn_gru(const float* __restrict__ agg,
                         float* __restrict__ h,
                         const float* __restrict__ w_ih,
                         const float* __restrict__ w_hh,
                         const float* __restrict__ b_ih,
                         const float* __restrict__ b_hh, int N) {
    __shared__ float s_wih[3 * CCH * CCH];
    __shared__ float s_whh[3 * CCH * CCH];
    __shared__ float s_bih[3 * CCH];
    __shared__ float s_bhh[3 * CCH];

    for (int i = threadIdx.x; i < 3 * CCH * CCH; i += blockDim.x) {
        s_wih[i] = w_ih[i];
        s_whh[i] = w_hh[i];
    }
    if (threadIdx.x < 3 * CCH) {
        s_bih[threadIdx.x] = b_ih[threadIdx.x];
        s_bhh[threadIdx.x] = b_hh[threadIdx.x];
    }
    __syncthreads();

    const int n = blockIdx.x * blockDim.x + threadIdx.x;
    if (n >= N) return;

    float a[CCH], hv[CCH];
    #pragma unroll
    for (int k = 0; k < CCH; ++k) {
        a[k]  = agg[(size_t)n * CCH + k];
        hv[k] = h  [(size_t)n * CCH + k];
    }

    #pragma unroll 1
    for (int j = 0; j < CCH; ++j) {
        float ir = s_bih[j], iz = s_bih[CCH + j], inn = s_bih[2 * CCH + j];
        float hr = s_bhh[j], hz = s_bhh[CCH + j], hnn = s_bhh[2 * CCH + j];
        #pragma unroll
        for (int k = 0; k < CCH; ++k) {
            ir  += a[k]  * s_wih[j * CCH + k];
            iz  += a[k]  * s_wih[(CCH + j) * CCH + k];
            inn += a[k]  * s_wih[(2 * CCH + j) * CCH + k];
            hr  += hv[k] * s_whh[j * CCH + k];
            hz  += hv[k] * s_whh[(CCH + j) * CCH + k];
            hnn += hv[k] * s_whh[(2 * CCH + j) * CCH + k];
        }
        const float r  = sigmoid_f(ir + hr);
        const float z  = sigmoid_f(iz + hz);
        const float nn = tanhf(inn + r * hnn);
        h[(size_t)n * CCH + j] = (1.0f - z) * nn + z * hv[j];
    }
}

// ---------------------------------------------------------------------------
// out = relu(h) @ lin_w.T + lin_b, thread per node, lin_w in LDS.
// ---------------------------------------------------------------------------
__global__ void ggnn_final(const float* __restrict__ h,
                           const float* __restrict__ lin_w,
                           const float* __restrict__ lin_b,
                           float* __restrict__ out, int N) {
    __shared__ float s_w[CCH * CCH];
    __shared__ float s_b[CCH];
    for (int i = threadIdx.x; i < CCH * CCH; i += blockDim.x) s_w[i] = lin_w[i];
    if (threadIdx.x < CCH) s_b[threadIdx.x] = lin_b[threadIdx.x];
    __syncthreads();

    const int n = blockIdx.x * blockDim.x + threadIdx.x;
    if (n >= N) return;

    float hv[CCH];
    #pragma unroll
    for (int k = 0; k < CCH; ++k) {
        const float v = h[(size_t)n * CCH + k];
        hv[k] = v > 0.0f ? v : 0.0f;
    }
    #pragma unroll 1
    for (int j = 0; j < CCH; ++j) {
        float acc = s_b[j];
        #pragma unroll
        for (int k = 0; k < CCH; ++k) acc += hv[k] * s_w[j * CCH + k];
        out[(size_t)n * CCH + j] = acc;
    }
}

// ---------------------------------------------------------------------------
extern "C" void kernel_launch(void* const* d_in, const int* in_sizes, int n_in,
                              void* d_out, int out_size, void* d_ws, size_t ws_size,
                              hipStream_t stream) {
    const float* x      = (const float*)d_in[0];
    const int*   eidx   = (const int*)  d_in[1];   // [2, E]
    const float* weight = (const float*)d_in[2];   // [3, 30, 30]
    const float* w_ih   = (const float*)d_in[3];   // [90, 30]
    const float* w_hh   = (const float*)d_in[4];   // [90, 30]
    const float* b_ih   = (const float*)d_in[5];   // [90]
    const float* b_hh   = (const float*)d_in[6];   // [90]
    const float* lin_w  = (const float*)d_in[7];   // [30, 30]
    const float* lin_b  = (const float*)d_in[8];   // [30]
    float* out = (float*)d_out;

    const int N = in_sizes[0] / CCH;
    const int E = in_sizes[1] / 2;
    const int* src = eidx;
    const int* dst = eidx + E;

    float* h   = (float*)d_ws;                      // N*30 f32
    float* m   = h + (size_t)N * CCH;               // N*30 f32
    float* agg = m + (size_t)N * CCH;               // N*30 f32

    const int nelem = N * CCH;
    const int n4    = (nelem >> 2) + 4;             // float4 threads (+tail)
    ggnn_copy<<<(n4 + 255) / 256, 256, 0, stream>>>(x, h, nelem);

    const int tiles = (N + 15) / 16;
    for (int layer = 0; layer < NLAYERS; ++layer) {
        ggnn_transform_wmma<<<(tiles + 3) / 4, 128, 0, stream>>>(
            h, weight + (size_t)layer * CCH * CCH, m, N);
        ggnn_zero<<<(n4 + 255) / 256, 256, 0, stream>>>(agg, nelem);
        ggnn_scatter<<<(E + 255) / 256, 256, 0, stream>>>(src, dst, m, agg, E);
        ggnn_gru<<<(N + 127) / 128, 128, 0, stream>>>(agg, h, w_ih, w_hh,
                                                      b_ih, b_hh, N);
    }
    ggnn_final<<<(N + 127) / 128, 128, 0, stream>>>(h, lin_w, lin_b, out, N);
}